// ODESYSLayer_17609365913954
// MI455X (gfx1250) — compile-verified
//
#include <hip/hip_runtime.h>
#include <hip/hip_bf16.h>

// ---- problem constants (from reference) ----
#define BSZ   128          // B = BS*NIND
#define NEQ   4
#define NSTEP 50
#define NDIM  4
#define OP1   3
#define MEQ   200          // NEQ*NSTEP
#define MIV   8
#define MD    392          // (NSTEP-1)*NDIM*ORDER
#define MC_   600          // MEQ+MIV+MD
#define NVAR_ 601
#define LD    608          // padded leading dim (38*16)
#define NT    38           // 16x16 tiles per side
#define NTRI  (NT*(NT+1)/2)   // 741 upper-triangular tiles

// workspace layout (floats, per batch)
#define A_OFF   ((size_t)0)
#define S_OFF   ((size_t)LD*LD)
#define B_OFF   ((size_t)2*LD*LD)
#define LAM_OFF (B_OFF + LD)
#define PERB    (LAM_OFF + LD)

// output layout (floats)
#define OUT_U0  0
#define OUT_U1  25600
#define OUT_U2  51200
#define OUT_EPS 76800
#define OUT_U   76928

typedef float v2f __attribute__((ext_vector_type(2)));
typedef float v8f __attribute__((ext_vector_type(8)));
typedef unsigned int u32x4 __attribute__((ext_vector_type(4)));
typedef int i32x4 __attribute__((ext_vector_type(4)));
typedef int i32x8 __attribute__((ext_vector_type(8)));

#if defined(__has_builtin)
#if __has_builtin(__builtin_amdgcn_tensor_load_to_lds) && \
    __has_builtin(__builtin_amdgcn_s_wait_tensorcnt)
#define HAVE_TDM 1
#endif
#endif

// ------------------------------------------------------------------
// K1: build dense padded A (608x608) and rhs vector b (608) per batch
// ------------------------------------------------------------------
__global__ __launch_bounds__(256)
void ode_build(const float* __restrict__ coeffs,  // (B,4,50,4,3)
               const float* __restrict__ rhs,     // (B,200)
               const float* __restrict__ ivr,     // (B,8)
               const float* __restrict__ steps,   // (B,49,4)
               float* __restrict__ ws) {
    const int b   = blockIdx.x;
    const int tid = threadIdx.x;
    float* A  = ws + (size_t)b * PERB + A_OFF;
    float* bv = ws + (size_t)b * PERB + B_OFF;

    // zero A (padded region must stay zero for WMMA tiles)
    for (int i = tid; i < LD * LD; i += 256) A[i] = 0.0f;
    __syncthreads();

    // EQ rows: 2400 entries
    for (int idx = tid; idx < NEQ * NSTEP * NDIM * OP1; idx += 256) {
        int o = idx % 3;
        int d = (idx / 3) % 4;
        int t = (idx / 12) % 50;
        int e = idx / 600;
        int row = e * NSTEP + t;
        int col = 1 + (t * NDIM + d) * OP1 + o;
        A[(size_t)row * LD + col] = coeffs[(size_t)b * 2400 + idx];
    }
    // IV rows: identity picks (8 entries)
    if (tid < MIV) {
        int di = tid >> 1, oi = tid & 1;
        A[(size_t)(MEQ + tid) * LD + (1 + di * OP1 + oi)] = 1.0f;
    }
    // D (Taylor continuity) rows: 392 rows
    for (int rid = tid; rid < MD; rid += 256) {
        int t = rid >> 3;          // /(NDIM*ORDER)
        int d = (rid >> 1) & 3;
        int i = rid & 1;
        int row = MEQ + MIV + rid;
        float h = steps[(size_t)b * ((NSTEP - 1) * NDIM) + t * NDIM + d];
        float* Ar = A + (size_t)row * LD;
        int cb = 1 + (t * NDIM + d) * OP1;
        for (int j = 0; j < OP1; ++j) {
            int pe = j - i;
            float v = 0.0f;
            if (pe == 0) v = -1.0f;
            else if (pe == 1) v = -h;
            else if (pe == 2) v = -0.5f * h * h;
            Ar[cb + j] = v;
        }
        Ar[1 + ((t + 1) * NDIM + d) * OP1 + i] = 1.0f;  // next-step coupling
        Ar[0] = -1.0f;                                   // eps column
    }
    // rhs vector
    for (int j = tid; j < LD; j += 256) {
        float v = 0.0f;
        if (j < MEQ)            v = rhs[(size_t)b * MEQ + j];
        else if (j < MEQ + MIV) v = ivr[(size_t)b * MIV + (j - MEQ)];
        bv[j] = v;
    }
}

// ------------------------------------------------------------------
// K2: S = (A*diag(hinv)) * A^T via V_WMMA_F32_16X16X4_F32
//     Upper-triangular tiles only (S symmetric; solver reads upper only).
//     One wave per 16x16 tile, K swept 8/iter with 2 WMMA accumulators.
// ------------------------------------------------------------------
__global__ __launch_bounds__(32)
void ode_form_S(float* __restrict__ ws) {
    const int lane = threadIdx.x;          // 0..31, wave32
    const int half = lane >> 4;            // 0/1
    const int l15  = lane & 15;

    // decode linear upper-triangular tile index -> (tm, tk), tk >= tm
    int L = blockIdx.x;
    int tm = 0, rowlen = NT;
    while (L >= rowlen) { L -= rowlen; --rowlen; ++tm; }
    const int tk = tm + L;
    const int m0 = tm * 16;
    const int k0 = tk * 16;

    const float* A = ws + (size_t)blockIdx.z * PERB + A_OFF;
    float*       S = ws + (size_t)blockIdx.z * PERB + S_OFF;

    // A-frag (16x4 f32): lane holds A[m0+l15][n0+2*half + {0,1}]  (scaled by hinv)
    // B-frag (4x16 f32): lane holds A[k0+l15][n0+2*half + {0,1}]  (A^T columns)
    const float* pa = A + (size_t)(m0 + l15) * LD + 2 * half;
    const float* pb = A + (size_t)(k0 + l15) * LD + 2 * half;

    v8f c0 = {}; v8f c1 = {};
    for (int n0 = 0; n0 < LD; n0 += 8) {
        __builtin_prefetch(pa + n0 + 64, 0, 0);
        __builtin_prefetch(pb + n0 + 64, 0, 0);
        v2f a0 = *(const v2f*)(pa + n0);
        v2f b0 = *(const v2f*)(pb + n0);
        v2f a1 = *(const v2f*)(pa + n0 + 4);
        v2f b1 = *(const v2f*)(pb + n0 + 4);
        // hinv: col 0 -> 1/GAMMA = 2, all others -> 1/ALPHA = 10
        float hv0 = (n0 == 0 && half == 0) ? 2.0f : 10.0f;
        a0.x *= hv0;   a0.y *= 10.0f;
        a1.x *= 10.0f; a1.y *= 10.0f;
        c0 = __builtin_amdgcn_wmma_f32_16x16x4_f32(false, a0, false, b0,
                                                   (short)0, c0, false, false);
        c1 = __builtin_amdgcn_wmma_f32_16x16x4_f32(false, a1, false, b1,
                                                   (short)0, c1, false, false);
    }
    // C/D 16x16 f32 layout: VGPR v, lanes 0-15 -> M=v, lanes 16-31 -> M=v+8
    const int col = k0 + l15;
#pragma unroll
    for (int v = 0; v < 8; ++v) {
        int row = m0 + half * 8 + v;
        S[(size_t)row * LD + col] = c0[v] + c1[v];
    }
}

// ------------------------------------------------------------------
// K3: in-place upper Cholesky S = U^T U (600x600), then
//     forward solve U^T y = b, backward solve U lam = y.
//     Row k of U is staged in LDS so the trailing rank-1 update
//     (the runtime dominator) reads both operands from LDS.
// ------------------------------------------------------------------
__global__ __launch_bounds__(256)
void ode_factor_solve(float* __restrict__ ws) {
    const int b    = blockIdx.x;
    const int tid  = threadIdx.x;
    const int wid  = tid >> 5;
    const int lane = tid & 31;
    float* S  = ws + (size_t)b * PERB + S_OFF;
    float* bv = ws + (size_t)b * PERB + B_OFF;
    float* lg = ws + (size_t)b * PERB + LAM_OFF;

    __shared__ float s_row[MC_];     // scaled pivot row U[k][*]
    __shared__ float s_r[MC_];
    __shared__ float s_lam[MC_];
    __shared__ float s_red[256];
    __shared__ float s_val;

    // ---- Cholesky (right-looking, upper) ----
    for (int k = 0; k < MC_; ++k) {
        if (tid == 0) {
            float dv = sqrtf(S[(size_t)k * LD + k]);
            S[(size_t)k * LD + k] = dv;
            s_val = 1.0f / dv;
        }
        __syncthreads();
        float invd = s_val;
        // scale pivot row; stash scaled row in LDS for the trailing update
        for (int j = k + 1 + tid; j < MC_; j += 256) {
            float u = S[(size_t)k * LD + j] * invd;
            S[(size_t)k * LD + j] = u;
            s_row[j] = u;
        }
        __syncthreads();
        // trailing update: wave per row i, lanes sweep columns j >= i
        for (int i = k + 1 + wid; i < MC_; i += 8) {
            float uki = s_row[i];
            for (int j = i + lane; j < MC_; j += 32)
                S[(size_t)i * LD + j] -= uki * s_row[j];
        }
        __syncthreads();
    }

    // ---- forward: U^T y = b (column sweep) ----
    for (int j = tid; j < MC_; j += 256) s_r[j] = bv[j];
    __syncthreads();
    for (int k = 0; k < MC_; ++k) {
        if (tid == 0) s_r[k] /= S[(size_t)k * LD + k];
        __syncthreads();
        float yk = s_r[k];
        for (int j = k + 1 + tid; j < MC_; j += 256)
            s_r[j] -= S[(size_t)k * LD + j] * yk;
        __syncthreads();
    }

    // ---- backward: U lam = y (row dot + block reduction) ----
    for (int k = MC_ - 1; k >= 0; --k) {
        float p = 0.0f;
        for (int j = k + 1 + tid; j < MC_; j += 256)
            p += S[(size_t)k * LD + j] * s_lam[j];
        s_red[tid] = p;
        __syncthreads();
        for (int s = 128; s > 0; s >>= 1) {
            if (tid < s) s_red[tid] += s_red[tid + s];
            __syncthreads();
        }
        if (tid == 0)
            s_lam[k] = (s_r[k] - s_red[0]) / S[(size_t)k * LD + k];
        __syncthreads();
    }
    for (int j = tid; j < MC_; j += 256) lg[j] = s_lam[j];
}

// ------------------------------------------------------------------
// K4: x = hinv * (A^T lam), scatter into the 5 concatenated outputs.
//     lam (2.4KB) is pulled into LDS via the Tensor Data Mover.
// ------------------------------------------------------------------
__global__ __launch_bounds__(256)
void ode_x_out(const float* __restrict__ ws, float* __restrict__ out) {
    const int b   = blockIdx.x;
    const int tid = threadIdx.x;
    const float* A  = ws + (size_t)b * PERB + A_OFF;
    const float* lg = ws + (size_t)b * PERB + LAM_OFF;

    __shared__ float s_lam[MC_];

#if defined(HAVE_TDM)
    if (tid < 32) {   // wave 0 issues the DMA descriptor (TDM ignores EXEC)
        unsigned long long ga = (unsigned long long)(const void*)lg;
        unsigned ldsoff = (unsigned)(unsigned long long)(void*)&s_lam[0];
        // ---- D# group 0: count=1 | lds_addr | 57-bit global addr | type=2 ----
        u32x4 g0;
        g0[0] = 1u;                                   // count=1, user descriptor
        g0[1] = ldsoff;                               // lds_addr (bytes)
        g0[2] = (unsigned)ga;                         // global_addr[31:0]
        g0[3] = (unsigned)((ga >> 32) & 0x01FFFFFFull) | (2u << 30); // addr[56:32], type=2
        // ---- D# group 1: 1 row x 600 f32, dense ----
        i32x8 g1;
        const unsigned dim0 = MC_;                    // 600 elements
        g1[0] = 0x20000;                              // wg_mask=0, data_size=2 (4B)
        g1[1] = (int)((dim0 & 0xFFFFu) << 16);        // tensor_dim0[15:0] (bits 63:48)
        g1[2] = (int)(((dim0 >> 16) & 0xFFFFu)        // tensor_dim0[31:16]
                      | (1u << 16));                  // tensor_dim1[15:0] = 1
        g1[3] = (int)((dim0 & 0xFFFFu) << 16);        // dim1 hi=0, tile_dim0=600
        g1[4] = 1;                                    // tile_dim1=1, tile_dim2=0
        g1[5] = (int)dim0;                            // tensor_dim0_stride lo
        g1[6] = 0;
        g1[7] = 0;
        i32x4 gz4 = {0, 0, 0, 0};                     // groups 2/3 unused (<=2D)
        i32x8 gz8 = {0, 0, 0, 0, 0, 0, 0, 0};         // extra group (clang-23 6-arg form)
        __builtin_amdgcn_tensor_load_to_lds(g0, g1, gz4, gz4, gz8, 0);
        __builtin_amdgcn_s_wait_tensorcnt(0);
    }
    __syncthreads();
#else
    for (int j = tid; j < MC_; j += 256) s_lam[j] = lg[j];
    __syncthreads();
#endif

    for (int n = tid; n < NVAR_; n += 256) {
        float acc = 0.0f;
        for (int m = 0; m < MC_; ++m)
            acc += A[(size_t)m * LD + n] * s_lam[m];   // coalesced across threads
        float x = ((n == 0) ? 2.0f : 10.0f) * acc;     // hinv
        if (n == 0) {
            out[OUT_EPS + b] = x;
        } else {
            int idx = n - 1;
            int o  = idx % 3;
            int td = idx / 3;                           // t*NDIM+d, 0..199
            out[(o == 0 ? OUT_U0 : o == 1 ? OUT_U1 : OUT_U2) + b * 200 + td] = x;
            out[OUT_U + (size_t)b * 600 + idx] = x;
        }
    }
}

// ------------------------------------------------------------------
extern "C" void kernel_launch(void* const* d_in, const int* in_sizes, int n_in,
                              void* d_out, int out_size, void* d_ws, size_t ws_size,
                              hipStream_t stream) {
    const float* coeffs = (const float*)d_in[0];
    const float* rhs    = (const float*)d_in[1];
    const float* ivr    = (const float*)d_in[2];
    const float* steps  = (const float*)d_in[3];
    float* out = (float*)d_out;
    float* ws  = (float*)d_ws;

    ode_build<<<dim3(BSZ), dim3(256), 0, stream>>>(coeffs, rhs, ivr, steps, ws);
    ode_form_S<<<dim3(NTRI, 1, BSZ), dim3(32), 0, stream>>>(ws);
    ode_factor_solve<<<dim3(BSZ), dim3(256), 0, stream>>>(ws);
    ode_x_out<<<dim3(BSZ), dim3(256), 0, stream>>>(ws, out);
}